// SplineconvModel_36988258353249
// MI455X (gfx1250) — compile-verified
//
#include <hip/hip_runtime.h>

// SplineConv (dim=1, degree=1, K=2) two-layer GNN for MI455X (gfx1250).
// Strategy: materialize Z = x@W[k] (51.2 MB -- L2 resident on the 192MB L2)
// with f32 WMMA, then a memory-bound edge gather/blend/atomic-scatter phase
// that runs out of L2, then mean+root+ReLU finalize.

typedef float v2f __attribute__((ext_vector_type(2)));
typedef float v8f __attribute__((ext_vector_type(8)));

#define NNODES 100000
#define DCH    64
#define NKER   2
#define NEDGE  1000000

// ---------------------------------------------------------------------------
// Zero scratch (agg + cnt are contiguous)
// ---------------------------------------------------------------------------
__global__ void __launch_bounds__(256) zero_kernel(float* __restrict__ p, int n) {
    int i = blockIdx.x * 256 + threadIdx.x;
    if (i < n) p[i] = 0.0f;
}

// ---------------------------------------------------------------------------
// Fused GEMM: one wave computes a 16x16 tile of the [N, 192] concatenated
// output  [ x@W[0] | x@W[1] | x@lw + b ]  using V_WMMA_F32_16X16X4_F32,
// chaining 16 WMMAs over the 64-deep reduction.
//   cols   0..127 -> Zc[n*128 + col]      (col = k*64 + o)
//   cols 128..191 -> R[n*64 + (col-128)] + b
// A layout (16x4 f32): lanes 0-15 hold K={k0,k0+1}, lanes 16-31 K={k0+2,k0+3},
// row M = lane&15.  B layout mirrored.  C/D: VGPR j -> M = j + 8*(lane>=16),
// N = lane&15.
// ---------------------------------------------------------------------------
__global__ void __launch_bounds__(256) spline_gemm(const float* __restrict__ X,
                                                   const float* __restrict__ W,   // [2,64,64] (k,in,out)
                                                   const float* __restrict__ LW,  // [64,64]   (in,out)
                                                   const float* __restrict__ B,   // [64]
                                                   float* __restrict__ Zc,        // [N,128]
                                                   float* __restrict__ R)         // [N,64]
{
    const int wave  = (blockIdx.x * 256 + threadIdx.x) >> 5;
    const int lane  = threadIdx.x & 31;
    const int mtile = wave / 12;          // 6250 row tiles
    const int ntile = wave % 12;          // 12 col tiles (192 cols)
    if (mtile >= NNODES / 16) return;     // wave-uniform: EXEC stays all-ones

    const int half = lane >> 4;           // 0: lanes 0-15, 1: lanes 16-31
    const int r    = lane & 15;

    // Select which weight matrix this col-tile reads (uniform per wave).
    const float* Bm;
    int col;
    if (ntile < 4)      { Bm = W;              col = ntile * 16 + r; }
    else if (ntile < 8) { Bm = W + DCH * DCH;  col = (ntile - 4) * 16 + r; }
    else                { Bm = LW;             col = (ntile - 8) * 16 + r; }

    const float* xr = X + (size_t)(mtile * 16 + r) * DCH;

    v8f c = {};
#pragma unroll
    for (int k0 = 0; k0 < DCH; k0 += 4) {
        const int kk = k0 + 2 * half;
        v2f a, b;
        a.x = xr[kk];
        a.y = xr[kk + 1];
        b.x = Bm[(size_t)kk * DCH + col];
        b.y = Bm[(size_t)(kk + 1) * DCH + col];
        c = __builtin_amdgcn_wmma_f32_16x16x4_f32(
                /*neg_a=*/false, a, /*neg_b=*/false, b,
                /*c_mod=*/(short)0, c, /*reuse_a=*/false, /*reuse_b=*/false);
    }

    const int gcol = ntile * 16 + r;      // 0..191
#pragma unroll
    for (int j = 0; j < 8; ++j) {
        const int row = mtile * 16 + j + 8 * half;
        if (gcol < 128) {
            Zc[(size_t)row * 128 + gcol] = c[j];
        } else {
            const int oc = gcol - 128;
            R[(size_t)row * DCH + oc] = c[j] + B[oc];
        }
    }
}

// ---------------------------------------------------------------------------
// Edge phase: 16 lanes per edge, 4 channels each (float4 gathers from the
// L2-resident Zc), blend by frac, hardware f32 atomic scatter into agg.
// K=2 open B-spline: v = u, i0 = clamp(floor(v)) , i1 = min(i0+1, 1).
// ---------------------------------------------------------------------------
__global__ void __launch_bounds__(256) spline_edges(const int*   __restrict__ EI,   // [2,E]
                                                    const float* __restrict__ U,    // [E,1]
                                                    const float* __restrict__ Zc,   // [N,128]
                                                    float* __restrict__ agg,        // [N,64]
                                                    float* __restrict__ cnt)        // [N]
{
    const long long t = (long long)blockIdx.x * 256 + threadIdx.x;
    const int e  = (int)(t >> 4);
    const int cq = (int)(t & 15);
    if (e >= NEDGE) return;

    const int src = EI[e];
    const int dst = EI[NEDGE + e];

    const float u  = U[e];
    float v  = u * (float)(NKER - 1);
    float f0 = fminf(fmaxf(floorf(v), 0.0f), (float)(NKER - 1));
    const int i0 = (int)f0;
    const float frac = v - f0;
    const int i1 = min(i0 + 1, NKER - 1);

    const float4 z0 = *(const float4*)(Zc + (size_t)src * 128 + i0 * DCH + cq * 4);
    const float4 z1 = *(const float4*)(Zc + (size_t)src * 128 + i1 * DCH + cq * 4);
    const float w0 = 1.0f - frac, w1 = frac;

    float* a = agg + (size_t)dst * DCH + cq * 4;
    unsafeAtomicAdd(a + 0, w0 * z0.x + w1 * z1.x);
    unsafeAtomicAdd(a + 1, w0 * z0.y + w1 * z1.y);
    unsafeAtomicAdd(a + 2, w0 * z0.z + w1 * z1.z);
    unsafeAtomicAdd(a + 3, w0 * z0.w + w1 * z1.w);
    if (cq == 0) unsafeAtomicAdd(cnt + dst, 1.0f);
}

// ---------------------------------------------------------------------------
// out = relu(agg / max(cnt,1) + root)
// ---------------------------------------------------------------------------
__global__ void __launch_bounds__(256) spline_finalize(const float* __restrict__ agg,
                                                       const float* __restrict__ cnt,
                                                       const float* __restrict__ R,
                                                       float* __restrict__ out)
{
    const int i = blockIdx.x * 256 + threadIdx.x;
    if (i >= NNODES * DCH) return;
    const int n = i >> 6;
    const float c = fmaxf(cnt[n], 1.0f);
    out[i] = fmaxf(agg[i] / c + R[i], 0.0f);
}

// ---------------------------------------------------------------------------
extern "C" void kernel_launch(void* const* d_in, const int* in_sizes, int n_in,
                              void* d_out, int out_size, void* d_ws, size_t ws_size,
                              hipStream_t stream) {
    const float* x   = (const float*)d_in[0];
    const int*   ei  = (const int*)  d_in[1];
    const float* ea  = (const float*)d_in[2];
    const float* W1  = (const float*)d_in[3];
    const float* lw1 = (const float*)d_in[4];
    const float* b1  = (const float*)d_in[5];
    const float* W2  = (const float*)d_in[6];
    const float* lw2 = (const float*)d_in[7];
    const float* b2  = (const float*)d_in[8];
    float* out = (float*)d_out;

    // Workspace layout (floats): Zc[N*128] | R[N*64] | agg[N*64] | cnt[N]
    float* Zc  = (float*)d_ws;
    float* R   = Zc  + (size_t)NNODES * 128;
    float* agg = R   + (size_t)NNODES * DCH;
    float* cnt = agg + (size_t)NNODES * DCH;

    const int gemm_waves  = (NNODES / 16) * 12;                 // 75000
    const int gemm_blocks = gemm_waves / 8;                     // 9375 (exact)
    const int edge_blocks = (NEDGE * 16 + 255) / 256;           // 62500
    const int fin_blocks  = (NNODES * DCH + 255) / 256;
    const int zero_n      = NNODES * (DCH + 1);                 // agg + cnt contiguous
    const int zero_blocks = (zero_n + 255) / 256;

    // ---- layer 1 ----
    zero_kernel   <<<zero_blocks, 256, 0, stream>>>(agg, zero_n);
    spline_gemm   <<<gemm_blocks, 256, 0, stream>>>(x, W1, lw1, b1, Zc, R);
    spline_edges  <<<edge_blocks, 256, 0, stream>>>(ei, ea, Zc, agg, cnt);
    spline_finalize<<<fin_blocks, 256, 0, stream>>>(agg, cnt, R, out);   // out = h1

    // ---- layer 2 ----
    zero_kernel   <<<zero_blocks, 256, 0, stream>>>(agg, zero_n);
    spline_gemm   <<<gemm_blocks, 256, 0, stream>>>(out, W2, lw2, b2, Zc, R);
    spline_edges  <<<edge_blocks, 256, 0, stream>>>(ei, ea, Zc, agg, cnt);
    spline_finalize<<<fin_blocks, 256, 0, stream>>>(agg, cnt, R, out);   // final
}